// GradLoss_90237262889697
// MI455X (gfx1250) — compile-verified
//
#include <hip/hip_runtime.h>
#include <hip/hip_bf16.h>

// ---------------------------------------------------------------------------
// Sobel gradient loss, fused single pass over d = target - output.
//   loss = ( sum(sobelx(d)^2) + sum(sobely(d)^2) ) / N,  N = 32*512*512
// Memory-bound: ~67MB read @ 23.3 TB/s  ->  ~3us floor.
// ---------------------------------------------------------------------------

#define IMG_W   512
#define IMG_H   512
#define IMG_B   32
#define N_MEAN  (IMG_B * IMG_W * IMG_H)        // 8388608

#define TILE_W  64
#define TILE_H  16
#define HALO_W  (TILE_W + 2)                   // 66
#define HALO_H  (TILE_H + 2)                   // 18
#define NTHREADS 256                           // (64,4) -> 8 waves of 32

#define GRID_X  (IMG_W / TILE_W)               // 8
#define GRID_Y  (IMG_H / TILE_H)               // 32
#define NBLOCKS (GRID_X * GRID_Y * IMG_B)      // 8192

typedef __attribute__((ext_vector_type(2))) float v2f;
typedef __attribute__((ext_vector_type(8))) float v8f;

// ---------------------------------------------------------------------------
// Pass 1: fused diff + Sobel-X/Y stencil + block reduction -> partial sums.
// mode 0: write partial to ws[block]; mode 1: atomicAdd(out, partial/N).
// ---------------------------------------------------------------------------
__global__ __launch_bounds__(NTHREADS)
void sobel_loss_partials(const float* __restrict__ out_img,
                         const float* __restrict__ tgt_img,
                         float* __restrict__ partials,
                         float* __restrict__ result_atomic,
                         int mode) {
    __shared__ float tile[HALO_H][HALO_W];

    const int b  = blockIdx.z;
    const int x0 = blockIdx.x * TILE_W;
    const int y0 = blockIdx.y * TILE_H;
    const int tid = threadIdx.y * TILE_W + threadIdx.x;

    const float* __restrict__ o = out_img + (size_t)b * IMG_W * IMG_H;
    const float* __restrict__ t = tgt_img + (size_t)b * IMG_W * IMG_H;

    // Load halo tile of d = target - output (zero padding outside image).
    for (int i = tid; i < HALO_W * HALO_H; i += NTHREADS) {
        const int lx = i % HALO_W;
        const int ly = i / HALO_W;
        const int gx = x0 - 1 + lx;
        const int gy = y0 - 1 + ly;
        float v = 0.0f;
        if ((unsigned)gx < (unsigned)IMG_W && (unsigned)gy < (unsigned)IMG_H) {
            const int idx = gy * IMG_W + gx;
            v = t[idx] - o[idx];
        }
        tile[ly][lx] = v;
    }
    __syncthreads();

    // Each thread computes 4 output rows of its column.
    float acc = 0.0f;
    const int sx = threadIdx.x + 1;
    #pragma unroll
    for (int r = 0; r < TILE_H / 4; ++r) {
        const int sy = (threadIdx.y + r * 4) + 1;
        const float a  = tile[sy - 1][sx - 1];
        const float bb = tile[sy - 1][sx    ];
        const float c  = tile[sy - 1][sx + 1];
        const float d0 = tile[sy    ][sx - 1];
        const float f  = tile[sy    ][sx + 1];
        const float g  = tile[sy + 1][sx - 1];
        const float h  = tile[sy + 1][sx    ];
        const float i2 = tile[sy + 1][sx + 1];
        // Cross-correlation with KX=[[-1,0,1],[-2,0,2],[-1,0,1]],
        //                        KY=[[1,2,1],[0,0,0],[-1,-2,-1]]
        const float gx = (c - a) + 2.0f * (f - d0) + (i2 - g);
        const float gy = (a + 2.0f * bb + c) - (g + 2.0f * h + i2);
        acc += gx * gx + gy * gy;
    }

    // Wave32 reduction.
    #pragma unroll
    for (int off = 16; off > 0; off >>= 1)
        acc += __shfl_xor(acc, off, 32);

    __shared__ float wsum[NTHREADS / 32];
    const int wave = tid >> 5;
    const int lane = tid & 31;
    if (lane == 0) wsum[wave] = acc;
    __syncthreads();

    if (tid == 0) {
        float s = 0.0f;
        #pragma unroll
        for (int w = 0; w < NTHREADS / 32; ++w) s += wsum[w];
        if (mode == 0) {
            const int blk = (blockIdx.z * gridDim.y + blockIdx.y) * gridDim.x
                          + blockIdx.x;
            partials[blk] = s;
        } else {
            atomicAdd(result_atomic, s * (1.0f / (float)N_MEAN));
        }
    }
}

// ---------------------------------------------------------------------------
// Pass 2: single wave32 reduces 8192 partials. The 32 lane accumulators are
// collapsed with V_WMMA_F32_16X16X4_F32 against a ones B-matrix:
//   A layout (16x4 f32): VGPR0 = K=0 (lanes 0-15) / K=2 (lanes 16-31),
//                        VGPR1 = K=1 / K=3  (set to 0).
//   => D[m][n] = p[m] + p[m+16] for every n.
//   Lane L holds D[r + (L<16?0:8)][L%16] in its 8 D VGPRs, so
//   sum_r d[r] over lane0 and lane16 covers all 32 partials exactly once.
// ---------------------------------------------------------------------------
__global__ __launch_bounds__(32)
void sobel_loss_final(const float* __restrict__ partials,
                      float* __restrict__ out) {
    const int lane = threadIdx.x;   // 0..31, one full wave; EXEC all ones

    float p = 0.0f;
    for (int i = lane; i < NBLOCKS; i += 32)   // fixed per-lane order
        p += partials[i];

    v2f a; a[0] = p;    a[1] = 0.0f;
    v2f bo; bo[0] = 1.0f; bo[1] = 1.0f;
    v8f c = {};
    v8f d = __builtin_amdgcn_wmma_f32_16x16x4_f32(
        /*neg_a=*/false, a, /*neg_b=*/false, bo,
        /*c_mod=*/(short)0, c, /*reuse_a=*/false, /*reuse_b=*/false);

    float t = 0.0f;
    #pragma unroll
    for (int r = 0; r < 8; ++r) t += d[r];

    // rows 0..7 summed on lanes 0-15, rows 8..15 on lanes 16-31
    const float total = __shfl(t, 0, 32) + __shfl(t, 16, 32);

    if (lane == 0)
        out[0] = total * (1.0f / (float)N_MEAN);
}

__global__ void zero_scalar(float* p) { if (threadIdx.x == 0) p[0] = 0.0f; }

// ---------------------------------------------------------------------------
extern "C" void kernel_launch(void* const* d_in, const int* in_sizes, int n_in,
                              void* d_out, int out_size, void* d_ws, size_t ws_size,
                              hipStream_t stream) {
    const float* out_img = (const float*)d_in[0];   // "output"
    const float* tgt_img = (const float*)d_in[1];   // "target"
    float* res = (float*)d_out;
    float* ws  = (float*)d_ws;

    dim3 grid(GRID_X, GRID_Y, IMG_B);
    dim3 block(TILE_W, NTHREADS / TILE_W, 1);

    if (ws_size >= (size_t)NBLOCKS * sizeof(float)) {
        // Deterministic two-pass path.
        sobel_loss_partials<<<grid, block, 0, stream>>>(
            out_img, tgt_img, ws, res, /*mode=*/0);
        sobel_loss_final<<<1, 32, 0, stream>>>(ws, res);
    } else {
        // Fallback: atomic accumulation into the scalar.
        zero_scalar<<<1, 32, 0, stream>>>(res);
        sobel_loss_partials<<<grid, block, 0, stream>>>(
            out_img, tgt_img, ws, res, /*mode=*/1);
    }
}